// PairwiseSimilarityLoss_86079734546728
// MI455X (gfx1250) — compile-verified
//
#include <hip/hip_runtime.h>
#include <hip/hip_bf16.h>
#include <math.h>

typedef __attribute__((ext_vector_type(2))) float v2f;
typedef __attribute__((ext_vector_type(8))) float v8f;

#define N_ROWS 256
#define KCHUNKS 128          // K-split chunks per matrix
#define WAVES_PER_BLOCK 8    // 256 threads / wave32

// ---------------------------------------------------------------------------
// Gram kernel: G[mat] += X_tile * X_tile^T over this wave's K-chunk.
// Each wave computes a 64x64 macrotile as 4x4 WMMA 16x16 tiles (f32 16x16x4).
// grid = (16 macrotiles, KCHUNKS/WAVES_PER_BLOCK, 2 matrices), block = 256.
// ---------------------------------------------------------------------------
__global__ __launch_bounds__(256) void gram_kernel(
    const float* __restrict__ ada, const float* __restrict__ sou,
    float* __restrict__ gram, int D)
{
    const int wave  = (int)(threadIdx.x >> 5);
    const int lane  = (int)(threadIdx.x & 31);
    const int lrow  = lane & 15;   // row/col within 16-wide tile
    const int lhalf = lane >> 4;   // selects K pair {0,1} vs {2,3}

    const int tile = (int)blockIdx.x;          // 0..15
    const int mat  = (int)blockIdx.z;          // 0: ada, 1: sou
    const int row0 = (tile >> 2) * 64;
    const int col0 = (tile &  3) * 64;

    const float* __restrict__ X = (mat == 0) ? ada : sou;
    float* __restrict__ G = gram + (size_t)mat * N_ROWS * N_ROWS;

    const int kper = D / KCHUNKS;                           // 512 for D=65536
    const int chunk = (int)blockIdx.y * WAVES_PER_BLOCK + wave;
    const int kbeg = chunk * kper;
    const int kend = kbeg + kper;

    const v8f vzero = {0.f, 0.f, 0.f, 0.f, 0.f, 0.f, 0.f, 0.f};
    v8f acc[4][4];
#pragma unroll
    for (int i = 0; i < 4; ++i)
#pragma unroll
        for (int j = 0; j < 4; ++j)
            acc[i][j] = vzero;

    for (int kb = kbeg; kb < kend; kb += 4) {
        v2f afrag[4], bfrag[4];
#pragma unroll
        for (int g = 0; g < 4; ++g) {
            // A-matrix 16x4 f32 layout: lanes 0-15 hold K={kb,kb+1} of row
            // (row0+16g+lrow); lanes 16-31 hold K={kb+2,kb+3}.
            const float* pa =
                X + (size_t)(row0 + 16 * g + lrow) * D + kb + 2 * lhalf;
            afrag[g] = *(const v2f*)pa;
            // B-matrix 4x16 layout (B[k][n] = X[col0+n][k]): VGPR0/1 =
            // K rows {0,1} in lanes 0-15, {2,3} in lanes 16-31 -> same pattern.
            const float* pb =
                X + (size_t)(col0 + 16 * g + lrow) * D + kb + 2 * lhalf;
            bfrag[g] = *(const v2f*)pb;
        }
#pragma unroll
        for (int i = 0; i < 4; ++i)
#pragma unroll
            for (int j = 0; j < 4; ++j)
                acc[i][j] = __builtin_amdgcn_wmma_f32_16x16x4_f32(
                    false, afrag[i], false, bfrag[j],
                    (short)0, acc[i][j], false, false);
    }

    // C/D 16x16 f32 layout: VGPR r holds row (r + 8*lhalf), col = lrow.
#pragma unroll
    for (int i = 0; i < 4; ++i) {
        const int R = row0 + 16 * i;
#pragma unroll
        for (int j = 0; j < 4; ++j) {
            const int C = col0 + 16 * j;
#pragma unroll
            for (int r = 0; r < 8; ++r) {
                atomicAdd(&G[(size_t)(R + r + 8 * lhalf) * N_ROWS + C + lrow],
                          acc[i][j][r]);
            }
        }
    }
}

// ---------------------------------------------------------------------------
// inv_norm[i] = 1 / max(sqrt(G[i,i]), 1e-8)   (cosine-similarity eps)
// ---------------------------------------------------------------------------
__global__ void norm_kernel(const float* __restrict__ gram,
                            float* __restrict__ invn)
{
    const int i = (int)threadIdx.x;   // 256 threads
    for (int m = 0; m < 2; ++m) {
        float d = gram[(size_t)m * N_ROWS * N_ROWS + (size_t)i * N_ROWS + i];
        float nrm = sqrtf(fmaxf(d, 0.f));
        invn[m * N_ROWS + i] = 1.f / fmaxf(nrm, 1e-8f);
    }
}

// ---------------------------------------------------------------------------
// Per-row masked softmax for both matrices + KL term, reduced into out[0].
// ---------------------------------------------------------------------------
__device__ __forceinline__ float block_reduce(float v, float* red, bool do_max)
{
    const int t = (int)threadIdx.x;
    __syncthreads();
    red[t] = v;
    __syncthreads();
    for (int s = 128; s > 0; s >>= 1) {
        if (t < s)
            red[t] = do_max ? fmaxf(red[t], red[t + s]) : (red[t] + red[t + s]);
        __syncthreads();
    }
    return red[0];
}

__global__ __launch_bounds__(256) void kl_kernel(
    const float* __restrict__ gram, const float* __restrict__ invn,
    float* __restrict__ out)
{
    __shared__ float red[256];
    const int i = (int)blockIdx.x;
    const int j = (int)threadIdx.x;

    const float* Ga = gram;
    const float* Gs = gram + (size_t)N_ROWS * N_ROWS;
    const float* ia = invn;
    const float* is_ = invn + N_ROWS;

    const bool diag = (j == i);
    float sa = diag ? -INFINITY : Ga[(size_t)i * N_ROWS + j] * ia[i] * ia[j];
    float ss = diag ? -INFINITY : Gs[(size_t)i * N_ROWS + j] * is_[i] * is_[j];

    float ma = block_reduce(sa, red, true);
    float ms = block_reduce(ss, red, true);

    float ea = diag ? 0.f : expf(sa - ma);
    float es = diag ? 0.f : expf(ss - ms);

    float Sa = block_reduce(ea, red, false);
    float Ss = block_reduce(es, red, false);

    float pa = ea / Sa + 1e-8f;  // p_ada + EPS_KL
    float ps = es / Ss + 1e-8f;  // p_sou + EPS_KL
    float term = ps * (logf(ps) - logf(pa));   // diag: eps*(log e - log e)=0

    float tsum = block_reduce(term, red, false);
    if (j == 0)
        atomicAdd(out, tsum / (float)N_ROWS);
}

// ---------------------------------------------------------------------------
extern "C" void kernel_launch(void* const* d_in, const int* in_sizes, int n_in,
                              void* d_out, int out_size, void* d_ws,
                              size_t ws_size, hipStream_t stream)
{
    const float* ada = (const float*)d_in[0];
    const float* sou = (const float*)d_in[1];
    float* out  = (float*)d_out;
    float* gram = (float*)d_ws;                         // [2][256][256]
    float* invn = gram + 2 * N_ROWS * N_ROWS;           // [2][256]

    const int D = in_sizes[0] / N_ROWS;                 // 65536

    hipMemsetAsync(d_ws, 0,
                   (size_t)(2 * N_ROWS * N_ROWS + 2 * N_ROWS) * sizeof(float),
                   stream);
    hipMemsetAsync(d_out, 0, (size_t)out_size * sizeof(float), stream);

    dim3 ggrid(16, KCHUNKS / WAVES_PER_BLOCK, 2);       // (16,16,2) blocks
    gram_kernel<<<ggrid, 256, 0, stream>>>(ada, sou, gram, D);
    norm_kernel<<<1, 256, 0, stream>>>(gram, invn);
    kl_kernel<<<N_ROWS, 256, 0, stream>>>(gram, invn, out);
}